// TopoEncoder_50852412784911
// MI455X (gfx1250) — compile-verified
//
#include <hip/hip_runtime.h>
#include <cstdint>
#include <cstddef>

// GNN TopoEncoder for MI455X (gfx1250, wave32).
//
// Roofline: 4.9 GFLOP total (trivial) vs ~3.2GB gather + ~3.2GB f32-atomic
// scatter per layer, all L2-class traffic (x fits in the 192MB L2 and has
// ~32x reuse per row). HBM sees only the streams. So: keep x/y L2-resident,
// mark one-shot streams (edges, embeds, d_out) non-temporal, stream the edge
// lists with CDNA5 async global->LDS DMA (ASYNCcnt), and scatter with
// hardware global_atomic_add_f32 (no-return, STOREcnt).

#define D        256
#define LN_EPS   1e-5f
#define CHUNK    32      // edges per wave per stage

typedef float f32x4 __attribute__((ext_vector_type(4)));

__device__ __forceinline__ float wave_sum(float v) {
#pragma unroll
  for (int off = 16; off > 0; off >>= 1) v += __shfl_xor(v, off, 32);
  return v;
}

// One wave (32 lanes) per node row: lane handles 8 contiguous floats.
// Writes LayerNorm(embeds) to xout and zero-fills yzero (first scatter target).
__global__ __launch_bounds__(256) void ln_zero_kernel(
    const float* __restrict__ in, float* __restrict__ xout,
    float* __restrict__ yzero, int n_rows) {
  const int row  = (int)((blockIdx.x * blockDim.x + threadIdx.x) >> 5);
  const int lane = threadIdx.x & 31;
  if (row >= n_rows) return;

  const f32x4* ip = (const f32x4*)(in + (size_t)row * D) + (lane << 1);
  f32x4 a = __builtin_nontemporal_load(ip);      // embeds: read once, NT
  f32x4 b = __builtin_nontemporal_load(ip + 1);

  float s  = a.x + a.y + a.z + a.w + b.x + b.y + b.z + b.w;
  float sq = a.x*a.x + a.y*a.y + a.z*a.z + a.w*a.w
           + b.x*b.x + b.y*b.y + b.z*b.z + b.w*b.w;
  s  = wave_sum(s);
  sq = wave_sum(sq);

  const float mean = s * (1.0f / D);
  const float ve   = sq * (1.0f / D) - mean * mean + LN_EPS;
  float r = rsqrtf(ve);
  r = r * (1.5f - 0.5f * ve * r * r);   // one Newton step for accuracy

  a = (a - mean) * r;
  b = (b - mean) * r;

  f32x4* xp = (f32x4*)(xout + (size_t)row * D) + (lane << 1);
  xp[0] = a;                       // x: L2-resident (heavy reuse next)
  xp[1] = b;

  const f32x4 z = {0.f, 0.f, 0.f, 0.f};
  f32x4* yp = (f32x4*)(yzero + (size_t)row * D) + (lane << 1);
  yp[0] = z;                       // y: L2-resident (atomic RMW target)
  yp[1] = z;
}

// COO SpMM: y[row[e]] += vals[e] * x[col[e]].
// Per-wave: async-stage 32 edge triples (b128 DMA, NT) into a private LDS
// slot, copy to registers, then process from registers (readlane broadcasts
// -> scalar addressing) while the next chunk streams into the same slot.
__global__ __launch_bounds__(256) void spmm_kernel(
    const int* __restrict__ erow, const int* __restrict__ ecol,
    const float* __restrict__ eval, const float* __restrict__ x,
    float* __restrict__ y, int n_edges) {
  // per wave: rows[32] | cols[32] | vals[32] | pad[32]  = 512 bytes
  __shared__ int lds[8 * 128];

  const int lane    = threadIdx.x & 31;
  const int wave    = threadIdx.x >> 5;
  const int gwave   = (int)((blockIdx.x * blockDim.x + threadIdx.x) >> 5);
  const int nwaves  = (int)((gridDim.x * blockDim.x) >> 5);
  const int nchunks = (n_edges + CHUNK - 1) / CHUNK;
  if (gwave >= nchunks) return;

  int* wbase = &lds[wave * 128];
  // Low 32 bits of a generic LDS pointer == DS byte offset (shared aperture).
  const unsigned wbase_bytes = (unsigned)(uintptr_t)wbase;

  // One b128 async DMA stages the whole 32-edge chunk:
  // lanes 0-7 -> rows, 8-15 -> cols, 16-23 -> vals, 24-31 -> pad (wasted).
  auto stage = [&](int chunk) {
    // Guard LDS RAW/WAR vs the async engine: our ds_loads of the previous
    // chunk must have completed before the DMA may overwrite the slot.
    asm volatile("s_wait_dscnt 0" ::: "memory");
    const int sel = lane >> 3;          // 0:rows 1:cols 2:vals 3:pad
    const int q   = lane & 7;           // 16B quad within the 128B array
    int e0 = chunk * CHUNK + (q << 2);
    if (e0 > n_edges - 4) e0 = n_edges - 4;   // clamp (tail handled separately)
    const void* src = (sel == 0) ? (const void*)(erow + e0)
                    : (sel == 1) ? (const void*)(ecol + e0)
                                 : (const void*)(eval + e0);
    const unsigned dst = wbase_bytes + (unsigned)(sel << 7) + (unsigned)(q << 4);
    asm volatile("global_load_async_to_lds_b128 %0, %1, off th:TH_LOAD_NT"
                 :: "v"(dst), "v"((unsigned long long)(uintptr_t)src)
                 : "memory");
  };

  // Process one edge: broadcast triple to SGPRs, gather row, scatter atomics.
  auto body = [&](int j, int rr, int cc, int vv) {
    const int   row = __builtin_amdgcn_readlane(rr, j);
    const int   col = __builtin_amdgcn_readlane(cc, j);
    const float val = __int_as_float(__builtin_amdgcn_readlane(vv, j));
    const float* xp = x + (size_t)col * D + (lane << 3);
    f32x4 a = *(const f32x4*)xp;
    f32x4 b = *(const f32x4*)(xp + 4);
    float* yp = y + (size_t)row * D + (lane << 3);
    unsafeAtomicAdd(yp + 0, a.x * val);
    unsafeAtomicAdd(yp + 1, a.y * val);
    unsafeAtomicAdd(yp + 2, a.z * val);
    unsafeAtomicAdd(yp + 3, a.w * val);
    unsafeAtomicAdd(yp + 4, b.x * val);
    unsafeAtomicAdd(yp + 5, b.y * val);
    unsafeAtomicAdd(yp + 6, b.z * val);
    unsafeAtomicAdd(yp + 7, b.w * val);
  };

  int chunk = gwave;
  stage(chunk);
  asm volatile("s_wait_asynccnt 0" ::: "memory");
  int rr = wbase[lane];
  int cc = wbase[32 + lane];
  int vv = wbase[64 + lane];

  while (chunk < nchunks) {
    const int  next = chunk + nwaves;
    const bool hn   = next < nchunks;
    if (hn) stage(next);   // DMA next chunk while we process from registers

    const int base = chunk * CHUNK;
    int cnt = n_edges - base;
    if (cnt >= CHUNK) {
#pragma unroll 4
      for (int j = 0; j < CHUNK; ++j) body(j, rr, cc, vv);
    } else {
      // Tail: b128 staging may mis-map slots; use direct clamped loads.
      int e = base + lane;
      if (e >= n_edges) e = n_edges - 1;
      const int tr = erow[e];
      const int tc = ecol[e];
      const int tv = __float_as_int(eval[e]);
      for (int j = 0; j < cnt; ++j) body(j, tr, tc, tv);
    }

    if (hn) {
      asm volatile("s_wait_asynccnt 0" ::: "memory");
      rr = wbase[lane];
      cc = wbase[32 + lane];
      vv = wbase[64 + lane];
    }
    chunk = next;
  }
}

// out (= or +=) y (out is streaming-once -> NT); zero-fill xold, which
// becomes the next layer's atomic scatter target (keep resident).
__global__ __launch_bounds__(256) void accum_zero_kernel(
    float* __restrict__ out, const float* __restrict__ y,
    float* __restrict__ xold, int n4, int first) {
  const int stride = (int)(gridDim.x * blockDim.x);
  f32x4* o = (f32x4*)out;
  const f32x4* yy = (const f32x4*)y;
  f32x4* xz = (f32x4*)xold;
  const f32x4 z = {0.f, 0.f, 0.f, 0.f};
  for (int i = (int)(blockIdx.x * blockDim.x + threadIdx.x); i < n4; i += stride) {
    f32x4 v = yy[i];                        // y: stays resident (it is next x)
    if (!first) {
      f32x4 ov = __builtin_nontemporal_load(o + i);
      v += ov;
    }
    __builtin_nontemporal_store(v, o + i);
    xz[i] = z;
  }
}

extern "C" void kernel_launch(void* const* d_in, const int* in_sizes, int n_in,
                              void* d_out, int out_size, void* d_ws, size_t ws_size,
                              hipStream_t stream) {
  const int*   erow   = (const int*)d_in[0];
  const int*   ecol   = (const int*)d_in[1];
  const float* evals  = (const float*)d_in[2];
  const float* embeds = (const float*)d_in[3];
  const int n_edges = in_sizes[0];
  const int n_nodes = in_sizes[3] / D;

  float* out  = (float*)d_out;
  float* xbuf = (float*)d_ws;                        // 102.4 MB
  float* ybuf = xbuf + (size_t)n_nodes * D;          // 102.4 MB

  // x = LayerNorm(embeds); y = 0
  ln_zero_kernel<<<(n_nodes + 7) / 8, 256, 0, stream>>>(embeds, xbuf, ybuf, n_nodes);

  float* xcur = xbuf;
  float* ycur = ybuf;
  const int n4 = (n_nodes * D) / 4;
  for (int l = 0; l < 3; ++l) {
    spmm_kernel<<<2048, 256, 0, stream>>>(erow, ecol, evals, xcur, ycur, n_edges);
    // out (+)= y; zero old x so it can serve as next layer's scatter target.
    accum_zero_kernel<<<1024, 256, 0, stream>>>(out, ycur, xcur, n4, l == 0 ? 1 : 0);
    float* t = xcur; xcur = ycur; ycur = t;
  }
}